// CMPLoss_31791347925635
// MI455X (gfx1250) — compile-verified
//
#include <hip/hip_runtime.h>
#include <stdint.h>

// ---------------------------------------------------------------------------
// CMP loss on MI455X (gfx1250, wave32).
// HBM-bound masked row-reduction over a 256MB f32 matrix:
//   - denom row-sums on the matrix pipe: V_WMMA_F32_16X16X4_F32 vs all-ones B
//     (two interleaved accumulators to break the D->C chain)
//   - labels staged to LDS via global_load_async_to_lds_b128 + s_wait_asynccnt
//   - probs streamed once with non-temporal b64 loads
//   - has_any == (denom > 0): selected values satisfy p > t >= 0, so they are
//     strictly positive and the sum is zero iff nothing was selected.
// ---------------------------------------------------------------------------

#define EPSILON 1e-10f

constexpr int WAVE            = 32;
constexpr int WAVES_PER_BLOCK = 8;
constexpr int BLOCK           = WAVE * WAVES_PER_BLOCK; // 256 threads
constexpr int MAX_B           = 8192;                   // labels LDS = 32KB
constexpr int KCHUNKS         = 4;                      // column partitions

typedef float v2f __attribute__((ext_vector_type(2)));
typedef float v8f __attribute__((ext_vector_type(8)));
typedef int   v2i __attribute__((ext_vector_type(2)));

// CDNA5 async copy: global -> LDS, 16B/lane, tracked by ASYNCcnt.
__device__ __forceinline__ void async_load_b128_to_lds(uint32_t lds_off,
                                                       const void* gaddr) {
  asm volatile("global_load_async_to_lds_b128 %0, %1, off"
               :
               : "v"(lds_off), "v"(gaddr)
               : "memory");
}

__device__ __forceinline__ void wait_asynccnt_zero() {
  asm volatile("s_wait_asynccnt 0" ::: "memory");
}

// ---------------------------------------------------------------------------
// Kernel 1: p_true[i] = probs[i*B + labels[i]]
// ---------------------------------------------------------------------------
__global__ void cmp_ptrue_kernel(const float* __restrict__ probs,
                                 const int* __restrict__ labels,
                                 float* __restrict__ p_true, int B) {
  int i = blockIdx.x * blockDim.x + threadIdx.x;
  if (i < B) {
    int l = labels[i];
    p_true[i] = probs[(size_t)i * (size_t)B + (size_t)l];
  }
}

// ---------------------------------------------------------------------------
// Kernel 2: one wave per (16-row group, 2048-col chunk).
// WMMA A layout (16x4 f32): lane L serves row M = L&15; its 2 K-elements are
// columns {j0, j0+1} (lanes 0-15) or {j0+2, j0+3} (lanes 16-31).  With B all
// ones, every column of the 16x16 D accumulator equals the row sum, so lane 0
// holds rows 0..7 in acc[0..7] and lane 16 holds rows 8..15.
// ---------------------------------------------------------------------------
__global__ void cmp_row_wmma_kernel(const float* __restrict__ probs,
                                    const int* __restrict__ labels,
                                    const float* __restrict__ p_true,
                                    float* __restrict__ denom_part, int B) {
  __shared__ __align__(16) int lds_labels[MAX_B];

  const int tid = threadIdx.x;
  // Stage all B labels into LDS with async b128 copies (8 per thread).
  for (int j = tid * 4; j < B; j += BLOCK * 4) {
    async_load_b128_to_lds((uint32_t)(uintptr_t)&lds_labels[j], &labels[j]);
  }
  wait_asynccnt_zero();
  __syncthreads();

  const int wave = tid / WAVE;
  const int lane = tid % WAVE;
  const int wid  = blockIdx.x * WAVES_PER_BLOCK + wave;
  const int g = wid / KCHUNKS;   // 16-row group
  const int c = wid % KCHUNKS;   // column chunk
  if (g >= B / 16) return;

  const int m    = lane & 15;    // row within group served by this lane
  const int half = lane >> 4;    // selects K-pair {0,1} vs {2,3}
  const int row  = g * 16 + m;

  const float t  = p_true[row];      // lane-constant row threshold
  const int   li = lds_labels[row];  // lane-constant row label

  // Hoist the per-lane K-pair offset into the base pointers.
  const float* __restrict__ gbase = probs + (size_t)row * (size_t)B + 2 * half;
  const int*   __restrict__ lbase = lds_labels + 2 * half;

  const int ckw   = B / KCHUNKS;     // 2048 columns per chunk
  const int j_beg = c * ckw;
  const int j_end = j_beg + ckw;

  v8f acc0 = {};                     // two accumulators: break D->C chain
  v8f acc1 = {};
  v2f ones; ones[0] = 1.0f; ones[1] = 1.0f;

  for (int j0 = j_beg; j0 < j_end; j0 += 16) {
#pragma unroll
    for (int u = 0; u < 4; u++) {
      const int j = j0 + 4 * u;
      v2f p = __builtin_nontemporal_load((const v2f*)(gbase + j)); // NT stream
      v2i l = *(const v2i*)(lbase + j);                            // ds b64
      bool s0 = (p[0] > t) && (l[0] != li);
      bool s1 = (p[1] > t) && (l[1] != li);
      v2f a;
      a[0] = s0 ? p[0] : 0.0f;
      a[1] = s1 ? p[1] : 0.0f;
      if (u & 1) {
        acc1 = __builtin_amdgcn_wmma_f32_16x16x4_f32(
            false, a, false, ones, (short)0, acc1, false, false);
      } else {
        acc0 = __builtin_amdgcn_wmma_f32_16x16x4_f32(
            false, a, false, ones, (short)0, acc0, false, false);
      }
    }
  }

  const int pbase = (g * KCHUNKS + c) * 16;
  // Lanes 0..15 hold rows 0..7; lanes 16..31 hold rows 8..15.
  if (lane == 0) {
#pragma unroll
    for (int r = 0; r < 8; r++) denom_part[pbase + r] = acc0[r] + acc1[r];
  }
  if (lane == 16) {
#pragma unroll
    for (int r = 0; r < 8; r++) denom_part[pbase + 8 + r] = acc0[r] + acc1[r];
  }
}

// ---------------------------------------------------------------------------
// Kernel 3: combine K-chunk partials and reduce:
//   out = (1/B) * sum_i [denom_i > 0] * p_true[i] / (denom_i + eps)
// Single block -> deterministic.
// ---------------------------------------------------------------------------
__global__ void cmp_finalize_kernel(const float* __restrict__ p_true,
                                    const float* __restrict__ denom_part,
                                    float* __restrict__ out, int B) {
  __shared__ float red[BLOCK];
  float s = 0.0f;
  for (int i = threadIdx.x; i < B; i += BLOCK) {
    const int g = i >> 4, m = i & 15;
    float d = 0.0f;
#pragma unroll
    for (int c = 0; c < KCHUNKS; c++) {
      d += denom_part[(g * KCHUNKS + c) * 16 + m];
    }
    if (d > 0.0f) s += p_true[i] / (d + EPSILON);
  }
  red[threadIdx.x] = s;
  __syncthreads();
  for (int off = BLOCK / 2; off > 0; off >>= 1) {
    if (threadIdx.x < off) red[threadIdx.x] += red[threadIdx.x + off];
    __syncthreads();
  }
  if (threadIdx.x == 0) out[0] = red[0] / (float)B;
}

// ---------------------------------------------------------------------------
extern "C" void kernel_launch(void* const* d_in, const int* in_sizes, int n_in,
                              void* d_out, int out_size, void* d_ws,
                              size_t ws_size, hipStream_t stream) {
  const float* probs  = (const float*)d_in[0];
  const int*   labels = (const int*)d_in[1];   // integer input -> const int*
  const int B = in_sizes[1];                   // 8192

  float* p_true     = (float*)d_ws;            // [B]
  float* denom_part = p_true + B;              // [B * KCHUNKS]
  float* out        = (float*)d_out;           // scalar f32

  cmp_ptrue_kernel<<<(B + 255) / 256, 256, 0, stream>>>(probs, labels, p_true, B);

  const int nwaves  = (B / 16) * KCHUNKS;                                // 2048
  const int nblocks = (nwaves + WAVES_PER_BLOCK - 1) / WAVES_PER_BLOCK;  // 256
  cmp_row_wmma_kernel<<<nblocks, BLOCK, 0, stream>>>(probs, labels, p_true,
                                                     denom_part, B);

  cmp_finalize_kernel<<<1, BLOCK, 0, stream>>>(p_true, denom_part, out, B);
}